// dModel_89498528514161
// MI455X (gfx1250) — compile-verified
//
#include <hip/hip_runtime.h>

#define IMG   2048
#define TW    128         // tile output width
#define TH    64          // tile output height
#define LDS_W 136         // halo span rounded to float4: cols [tile_w-4, tile_w+132)
#define LDS_H 66          // rows [tile_h-1, tile_h+65)
#define NQ    (LDS_W / 4) // 34 float4 per LDS row
#define NITEMS (LDS_H * NQ) // 2244 float4 transfers per tile
#define NTHREADS 256

typedef __attribute__((ext_vector_type(4))) float f4;

__global__ __launch_bounds__(NTHREADS)
void conv3x3_async_kernel(const float* __restrict__ x,
                          const float* __restrict__ weight,
                          const float* __restrict__ den,
                          float* __restrict__ out)
{
    __shared__ __align__(16) float tile[LDS_H * LDS_W];   // 35,904 B

    const int tid    = threadIdx.x;
    const int tile_w = blockIdx.x * TW;
    const int tile_h = blockIdx.y * TH;
    const int img    = blockIdx.z;

    // ---- Effective 3x3 weights first: uniform scalar loads overlap staging ----
    // eff_w = weight * outer(kx,kx), kx = [den, 1, den]
    const float d0 = den[0];
    const float kr[3] = { d0, 1.0f, d0 };
    float W[3][3];
    #pragma unroll
    for (int i = 0; i < 3; ++i)
        #pragma unroll
        for (int j = 0; j < 3; ++j)
            W[i][j] = weight[i * 3 + j] * kr[i] * kr[j];

    const float* xin = x + ((size_t)img << 22);   // 2048*2048 = 1<<22
    // Low 32 bits of a generic pointer into LDS == the LDS byte offset (flat aperture).
    const unsigned lds_base = (unsigned)(size_t)(void*)&tile[0];

    // ---- Stage (TH+2)x(TW+2) halo tile into LDS via CDNA5 async copies ----
    for (int i = tid; i < NITEMS; i += NTHREADS) {
        const int lr   = i / NQ;              // 0..65
        const int lc4  = i - lr * NQ;         // 0..33
        const int grow = tile_h - 1 + lr;
        const int gcol = tile_w - 4 + lc4 * 4;
        const unsigned lds_addr = lds_base + (unsigned)(lr * (LDS_W * 4) + lc4 * 16);
        const bool inb = (grow >= 0) & (grow < IMG) & (gcol >= 0) & (gcol < IMG);
        if (inb) {
            const float* gp = xin + (size_t)grow * IMG + gcol;
            asm volatile("global_load_async_to_lds_b128 %0, %1, off"
                         :: "v"(lds_addr), "v"(gp) : "memory");
        } else {
            // zero-fill padding cells (disjoint bytes from async-written cells)
            *(f4*)&tile[lr * LDS_W + lc4 * 4] = (f4){0.f, 0.f, 0.f, 0.f};
        }
    }
    asm volatile("s_wait_asynccnt 0" ::: "memory");
    __syncthreads();

    // ---- Compute: 8 rows x 4 cols per thread, all inputs from LDS ----
    const int tx   = tid & 31;      // 32 col-groups * 4 cols = 128 (wave spans a row)
    const int ty   = tid >> 5;      // 8 row-groups * 8 rows = 64
    const int lcol = 4 * tx + 4;    // LDS col of first output col (16B aligned)
    const int r0   = 8 * ty;        // local output row base

    float acc[8][4];
    #pragma unroll
    for (int r = 0; r < 8; ++r)
        #pragma unroll
        for (int j = 0; j < 4; ++j) acc[r][j] = 0.f;

    // LDS row (r0+k) feeds output rows r in [k-2, k] with weight row (k - r).
    #pragma unroll
    for (int k = 0; k < 10; ++k) {
        const float* row = &tile[(r0 + k) * LDS_W + (lcol - 1)];
        float v[6];
        #pragma unroll
        for (int m = 0; m < 6; ++m) v[m] = row[m];
        #pragma unroll
        for (int r = 0; r < 8; ++r) {
            if (k >= r && k <= r + 2) {
                const int wr = k - r;
                #pragma unroll
                for (int j = 0; j < 4; ++j)
                    acc[r][j] = fmaf(W[wr][0], v[j],
                                fmaf(W[wr][1], v[j + 1],
                                fmaf(W[wr][2], v[j + 2], acc[r][j])));
            }
        }
    }

    // ---- Non-temporal b128 output stores (output is never re-read) ----
    float* o = out + ((size_t)img << 22) + (size_t)(tile_h + r0) * IMG + tile_w + 4 * tx;
    #pragma unroll
    for (int r = 0; r < 8; ++r) {
        f4 val = { acc[r][0], acc[r][1], acc[r][2], acc[r][3] };
        __builtin_nontemporal_store(val, (f4*)(o + (size_t)r * IMG));
    }
}

extern "C" void kernel_launch(void* const* d_in, const int* in_sizes, int n_in,
                              void* d_out, int out_size, void* d_ws, size_t ws_size,
                              hipStream_t stream) {
    const float* x      = (const float*)d_in[0];
    const float* weight = (const float*)d_in[1];
    const float* den    = (const float*)d_in[2];
    float* out          = (float*)d_out;

    dim3 grid(IMG / TW, IMG / TH, 16);  // 16 x 32 x 16 = 8192 blocks
    conv3x3_async_kernel<<<grid, NTHREADS, 0, stream>>>(x, weight, den, out);
}